// MaxLogits_84507776516438
// MI455X (gfx1250) — compile-verified
//
#include <hip/hip_runtime.h>
#include <cstdint>

// Problem constants (from reference): B=16, S=2048, F=128, C=16
#define T_TOKENS 32768
#define F_FINE   128
#define C_COARSE 16
#define BLOCKS   512
#define THREADS  256            // 8 waves of 32
#define WAVES_PER_BLOCK 8
#define TOTAL_WAVES (BLOCKS * WAVES_PER_BLOCK)   // 4096 -> 8 tokens per wave

typedef float v2f __attribute__((ext_vector_type(2)));
typedef float v8f __attribute__((ext_vector_type(8)));

// ---------------------------------------------------------------------------
// Kernel 1: per-token masked coarse max + log-softmax NLL, block partial sums.
// ---------------------------------------------------------------------------
__global__ void __launch_bounds__(THREADS)
maxlogits_nll_kernel(const float* __restrict__ logits,
                     const long long* __restrict__ labels,
                     const float* __restrict__ mask,     // [F,C] row-major
                     float* __restrict__ block_sums) {
    __shared__ __align__(16) float maskLDS[F_FINE * C_COARSE]; // 8 KB
    __shared__ unsigned membLDS[F_FINE];
    __shared__ float wsumLDS[WAVES_PER_BLOCK];

    const int tid  = threadIdx.x;
    const int lane = tid & 31;
    const int wv   = tid >> 5;

    // --- Stage mask matrix into LDS with CDNA5 async global->LDS copies. ---
    // 2048 floats = 8192 B; 256 lanes x 16 B x 2 rounds.
    #pragma unroll
    for (int i = 0; i < 2; ++i) {
        const int fidx = (tid + i * THREADS) * 4;          // float index, 16B aligned
        unsigned lds_off = (unsigned)(uintptr_t)(&maskLDS[fidx]);
        const float* gsrc = mask + fidx;
        asm volatile("global_load_async_to_lds_b128 %0, %1, off"
                     :: "v"(lds_off), "v"(gsrc) : "memory");
    }
    asm volatile("s_wait_asynccnt 0" ::: "memory");
    __syncthreads();

    // --- Membership bitmask per fine class: bit c set iff mask[f,c] > 0 ---
    if (tid < F_FINE) {
        unsigned m = 0u;
        #pragma unroll
        for (int c = 0; c < C_COARSE; ++c)
            if (maskLDS[tid * C_COARSE + c] > 0.0f) m |= (1u << c);
        membLDS[tid] = m;
    }
    __syncthreads();

    // Each lane owns 4 consecutive fine classes of whichever token the wave holds.
    const unsigned m0 = membLDS[lane * 4 + 0];
    const unsigned m1 = membLDS[lane * 4 + 1];
    const unsigned m2 = membLDS[lane * 4 + 2];
    const unsigned m3 = membLDS[lane * 4 + 3];

    const int gw = blockIdx.x * WAVES_PER_BLOCK + wv;  // global wave id
    float acc = 0.0f;

    for (int t = gw; t < T_TOKENS; t += TOTAL_WAVES) {
        const float4 v = *(const float4*)(logits + (size_t)t * F_FINE + lane * 4);

        // Per-lane masked candidates for each coarse class.
        float cm[C_COARSE];
        #pragma unroll
        for (int c = 0; c < C_COARSE; ++c) {
            float x = -__FLT_MAX__;
            x = ((m0 >> c) & 1u) ? fmaxf(x, v.x) : x;
            x = ((m1 >> c) & 1u) ? fmaxf(x, v.y) : x;
            x = ((m2 >> c) & 1u) ? fmaxf(x, v.z) : x;
            x = ((m3 >> c) & 1u) ? fmaxf(x, v.w) : x;
            cm[c] = x;
        }
        // Wave-wide butterfly max per class -> uniform across lanes.
        #pragma unroll
        for (int c = 0; c < C_COARSE; ++c) {
            float x = cm[c];
            x = fmaxf(x, __shfl_xor(x, 1, 32));
            x = fmaxf(x, __shfl_xor(x, 2, 32));
            x = fmaxf(x, __shfl_xor(x, 4, 32));
            x = fmaxf(x, __shfl_xor(x, 8, 32));
            x = fmaxf(x, __shfl_xor(x, 16, 32));
            cm[c] = x;
        }
        // log-softmax NLL at the label (uniform computation, no dynamic indexing).
        float mx = cm[0];
        #pragma unroll
        for (int c = 1; c < C_COARSE; ++c) mx = fmaxf(mx, cm[c]);
        const int lab = (int)labels[t];
        float se = 0.0f, picked = 0.0f;
        #pragma unroll
        for (int c = 0; c < C_COARSE; ++c) {
            se += __expf(cm[c] - mx);
            picked = (c == lab) ? cm[c] : picked;
        }
        acc += __logf(se) + mx - picked;
    }

    if (lane == 0) wsumLDS[wv] = acc;
    __syncthreads();
    if (tid == 0) {
        float s = 0.0f;
        #pragma unroll
        for (int w = 0; w < WAVES_PER_BLOCK; ++w) s += wsumLDS[w];  // fixed order
        block_sums[blockIdx.x] = s;
    }
}

// ---------------------------------------------------------------------------
// Kernel 2: deterministic final reduction of 512 partials via WMMA.
// A = ones(16x4), so D = A*B + C accumulates the sum of every element fed
// through B, independent of the B register layout.
// ---------------------------------------------------------------------------
__global__ void __launch_bounds__(32)
wmma_sum_kernel(const float* __restrict__ partials, float* __restrict__ out) {
    const int lane = threadIdx.x;   // exactly one wave32, EXEC all ones
    v2f a; a.x = 1.0f; a.y = 1.0f;
    v8f c = {0.f, 0.f, 0.f, 0.f, 0.f, 0.f, 0.f, 0.f};
    #pragma unroll
    for (int i = 0; i < BLOCKS; i += 64) {
        v2f b;
        b.x = partials[i + lane];
        b.y = partials[i + 32 + lane];
        c = __builtin_amdgcn_wmma_f32_16x16x4_f32(false, a, false, b,
                                                  (short)0, c, false, false);
    }
    // c[0]: lanes 0-15 hold D[0][n], lanes 16-31 hold D[8][n]; rows identical.
    float s = c[0];
    s += __shfl_xor(s, 1, 32);
    s += __shfl_xor(s, 2, 32);
    s += __shfl_xor(s, 4, 32);
    s += __shfl_xor(s, 8, 32);
    s += __shfl_xor(s, 16, 32);          // = 2 * total
    if (lane == 0) out[0] = s * (0.5f / (float)T_TOKENS);
}

extern "C" void kernel_launch(void* const* d_in, const int* in_sizes, int n_in,
                              void* d_out, int out_size, void* d_ws, size_t ws_size,
                              hipStream_t stream) {
    (void)in_sizes; (void)n_in; (void)out_size; (void)ws_size;
    const float*     logits = (const float*)d_in[0];
    const long long* labels = (const long long*)d_in[1];   // int64 labels
    const float*     mask   = (const float*)d_in[2];
    float* block_sums = (float*)d_ws;                      // 512 floats (2 KB)
    float* out        = (float*)d_out;

    maxlogits_nll_kernel<<<BLOCKS, THREADS, 0, stream>>>(logits, labels, mask, block_sums);
    wmma_sum_kernel<<<1, 32, 0, stream>>>(block_sums, out);
}